// EmbeddingD_47287589929191
// MI455X (gfx1250) — compile-verified
//
#include <hip/hip_runtime.h>
#include <hip/hip_bf16.h>

// ---------------------------------------------------------------------------
// GCN x3 (two layers each) + SE attention + 1x1 conv for MI455X (gfx1250).
// Dense-adjacency reformulation: GCN aggregation == Ahat @ (X @ W), all four
// GEMMs per graph type run on V_WMMA_F32_16X16X4_F32 (fp32 WMMA, reference
// precision). Each wave computes a 16x64 strip: one shared A fragment feeds
// 4 independent WMMA accumulator chains (amortized loads, hidden WMMA latency).
// All operands are zero-padded to 592 rows so the K-loop has no predication.
// ---------------------------------------------------------------------------

#define N_DRUGS 591
#define NP      592          // N padded to a multiple of 16 (37 tiles)
#define FDIM    1024
#define NEDGE   120000

typedef __attribute__((ext_vector_type(2))) float v2f;
typedef __attribute__((ext_vector_type(8))) float v8f;

// ---------------- workspace init (re-run every call: ws is poisoned) --------
__global__ void init_ws(float* __restrict__ ahat, float* __restrict__ deg,
                        float* __restrict__ pool) {
  int idx = blockIdx.x * blockDim.x + threadIdx.x;
  int stride = gridDim.x * blockDim.x;
  const int AH = 3 * NP * NP;
  for (int i = idx; i < AH; i += stride) ahat[i] = 0.0f;
  for (int i = idx; i < 3 * NP; i += stride) deg[i] = 1.0f;  // self-loop weight
  if (idx < 8) pool[idx] = 0.0f;
}

// ---------------- zero-padded copy of dm1 -> [NP][FDIM] ---------------------
__global__ void pad_dm1(const float* __restrict__ dm1, float* __restrict__ out) {
  int idx = blockIdx.x * blockDim.x + threadIdx.x;
  if (idx >= NP * FDIM) return;
  int n = idx >> 10;
  out[idx] = (n < N_DRUGS) ? dm1[idx] : 0.0f;
}

// ---------------- degree accumulation (scalar atomic per edge) --------------
__global__ void deg_accum(const int* __restrict__ eT, const float* __restrict__ wT,
                          const int* __restrict__ eS, const float* __restrict__ wS,
                          const int* __restrict__ eG, const float* __restrict__ wG,
                          float* __restrict__ deg) {
  int idx = blockIdx.x * blockDim.x + threadIdx.x;
  if (idx >= 3 * NEDGE) return;
  int t = idx / NEDGE, e = idx - t * NEDGE;
  const int*   ed = (t == 0) ? eT : ((t == 1) ? eS : eG);
  const float* w  = (t == 0) ? wT : ((t == 1) ? wS : wG);
  int dst = ed[NEDGE + e];     // edges[0] = src, edges[1] = dst
  atomicAdd(&deg[t * NP + dst], w[e]);
}

__global__ void compute_dinv(const float* __restrict__ deg, float* __restrict__ dinv) {
  int i = blockIdx.x * blockDim.x + threadIdx.x;
  if (i < 3 * NP) {
    float d = deg[i];
    dinv[i] = (d > 0.0f) ? rsqrtf(d) : 0.0f;
  }
}

// ---------------- dense normalized adjacency build --------------------------
__global__ void build_ahat(const int* __restrict__ eT, const float* __restrict__ wT,
                           const int* __restrict__ eS, const float* __restrict__ wS,
                           const int* __restrict__ eG, const float* __restrict__ wG,
                           const float* __restrict__ dinv, float* __restrict__ ahat) {
  const int PER = NEDGE + N_DRUGS;
  int idx = blockIdx.x * blockDim.x + threadIdx.x;
  if (idx >= 3 * PER) return;
  int t = idx / PER, r = idx - t * PER;
  const float* dv = dinv + t * NP;
  float* Ah = ahat + (size_t)t * NP * NP;
  if (r < NEDGE) {
    const int*   ed = (t == 0) ? eT : ((t == 1) ? eS : eG);
    const float* w  = (t == 0) ? wT : ((t == 1) ? wS : wG);
    int src = ed[r], dst = ed[NEDGE + r];
    atomicAdd(&Ah[(size_t)dst * NP + src], dv[src] * w[r] * dv[dst]);
  } else {
    int i = r - NEDGE;                              // self loop, weight 1
    atomicAdd(&Ah[(size_t)i * NP + i], dv[i] * dv[i]);
  }
}

// ---------------- fp32 WMMA GEMM: C[M,1024] = A[M,K] @ B[K,1024] ------------
// One wave computes a 16(M) x 64(N) strip: 4 accumulators share one A-frag.
// A-frag layout (ISA 7.12.2): lane<16 -> K{0,1}, lane>=16 -> K{2,3}, M=lane&15.
// B-frag symmetric; C/D: VGPR r holds rows r / r+8 for the two lane groups.
// All operands padded: no predication inside the K-loop, EXEC all-ones.
__global__ void gemm_wmma_f32(const float* __restrict__ A, int lda,
                              const float* __restrict__ B, int ldb,
                              float* __restrict__ C, int ldc,
                              int K, int n_wtiles, int m_tiles,
                              const float* __restrict__ bias, int do_relu) {
  int wid = (blockIdx.x * blockDim.x + threadIdx.x) >> 5;   // wave-uniform
  if (wid >= m_tiles * n_wtiles) return;                    // whole-wave exit
  int lane = threadIdx.x & 31;
  int tm = wid / n_wtiles, tn = wid - tm * n_wtiles;
  int row0 = tm << 4, col0 = tn << 6;                       // 64 cols per wave
  int ml = lane & 15;
  int khalf = (lane >> 4) << 1;                             // 0 or 2

  const float* Arow = A + (size_t)(row0 + ml) * lda + khalf;
  const float* B0   = B + (size_t)khalf * ldb + (col0 + ml);

  v8f acc0 = {}, acc1 = {}, acc2 = {}, acc3 = {};
#pragma unroll 2
  for (int kb = 0; kb < K; kb += 4) {
    v2f a;
    a.x = Arow[kb];                                         // contiguous -> b64
    a.y = Arow[kb + 1];
    const float* Bk = B0 + (size_t)kb * ldb;
    v2f b0, b1, b2, b3;
    b0.x = Bk[0];   b0.y = Bk[ldb];
    b1.x = Bk[16];  b1.y = Bk[ldb + 16];
    b2.x = Bk[32];  b2.y = Bk[ldb + 32];
    b3.x = Bk[48];  b3.y = Bk[ldb + 48];
    // (neg_a, A, neg_b, B, c_mod, C, reuse_a, reuse_b)
    acc0 = __builtin_amdgcn_wmma_f32_16x16x4_f32(false, a, false, b0, (short)0, acc0, false, false);
    acc1 = __builtin_amdgcn_wmma_f32_16x16x4_f32(false, a, false, b1, (short)0, acc1, false, false);
    acc2 = __builtin_amdgcn_wmma_f32_16x16x4_f32(false, a, false, b2, (short)0, acc2, false, false);
    acc3 = __builtin_amdgcn_wmma_f32_16x16x4_f32(false, a, false, b3, (short)0, acc3, false, false);
  }

  int crow0 = row0 + ((lane >> 4) << 3);                    // lanes 16-31 -> +8
  v8f accs[4] = {acc0, acc1, acc2, acc3};
#pragma unroll
  for (int g = 0; g < 4; ++g) {
    int ccol = col0 + (g << 4) + ml;
    float bv = (bias != nullptr) ? bias[ccol] : 0.0f;
#pragma unroll
    for (int r = 0; r < 8; ++r) {
      float v = accs[g][r] + bv;
      if (do_relu) v = fmaxf(v, 0.0f);
      C[(size_t)(crow0 + r) * ldc + ccol] = v;
    }
  }
}

// ---------------- channel mean pool (rows 0..590 only) ----------------------
__global__ void pool_reduce(const float* __restrict__ H, float* __restrict__ pool) {
  int c = blockIdx.y;
  const float* h = H + (size_t)c * NP * FDIM;   // first 591*1024 elems = rows 0..590
  const int TOT = N_DRUGS * FDIM;
  float s = 0.0f;
  for (int i = blockIdx.x * blockDim.x + threadIdx.x; i < TOT;
       i += gridDim.x * blockDim.x)
    s += h[i];
  __shared__ float red[256];
  int tid = threadIdx.x;
  red[tid] = s;
  __syncthreads();
  for (int off = 128; off > 0; off >>= 1) {
    if (tid < off) red[tid] += red[tid + off];
    __syncthreads();
  }
  if (tid == 0) atomicAdd(&pool[c], red[0]);
}

// ---------------- SE attention: 6 -> 30 -> 6 (single tiny block) ------------
__global__ void attn_kernel(const float* __restrict__ pool,
                            const float* __restrict__ fc1_w, const float* __restrict__ fc1_b,
                            const float* __restrict__ fc2_w, const float* __restrict__ fc2_b,
                            float* __restrict__ a_out) {
  __shared__ float p[6], h[30];
  int tid = threadIdx.x;
  if (tid < 6) p[tid] = pool[tid] * (1.0f / (591.0f * 1024.0f));
  __syncthreads();
  if (tid < 30) {
    float s = fc1_b[tid];
    for (int c = 0; c < 6; ++c) s += p[c] * fc1_w[tid * 6 + c];   // fc1_w [30,6]
    h[tid] = fmaxf(s, 0.0f);
  }
  __syncthreads();
  if (tid < 6) {
    float s = fc2_b[tid];
    for (int j = 0; j < 30; ++j) s += h[j] * fc2_w[tid * 30 + j]; // fc2_w [6,30]
    a_out[tid] = 1.0f / (1.0f + expf(-s));
  }
}

// ---------------- fused attention-scale + relu + 1x1 conv + transpose -------
__global__ void final_kernel(const float* __restrict__ H, const float* __restrict__ a,
                             const float* __restrict__ conv_w,
                             const float* __restrict__ conv_b,
                             float* __restrict__ out) {
  int idx = blockIdx.x * blockDim.x + threadIdx.x;
  if (idx >= N_DRUGS * FDIM) return;
  int n = idx >> 10, f = idx & (FDIM - 1);
  float s = conv_b[0];
#pragma unroll
  for (int c = 0; c < 6; ++c) {
    float v = a[c] * H[(size_t)c * NP * FDIM + (size_t)n * FDIM + f];
    s += conv_w[c] * fmaxf(v, 0.0f);
  }
  out[idx] = s;   // out is [N, FD] row-major
}

// ---------------------------------------------------------------------------
extern "C" void kernel_launch(void* const* d_in, const int* in_sizes, int n_in,
                              void* d_out, int out_size, void* d_ws, size_t ws_size,
                              hipStream_t stream) {
  (void)in_sizes; (void)n_in; (void)out_size; (void)ws_size;

  const float* dm1 = (const float*)d_in[0];
  const int*   edges[3] = {(const int*)d_in[1], (const int*)d_in[3], (const int*)d_in[5]};
  const float* ew[3]    = {(const float*)d_in[2], (const float*)d_in[4], (const float*)d_in[6]};
  const float* W1[3]    = {(const float*)d_in[7],  (const float*)d_in[11], (const float*)d_in[15]};
  const float* b1[3]    = {(const float*)d_in[8],  (const float*)d_in[12], (const float*)d_in[16]};
  const float* W2[3]    = {(const float*)d_in[9],  (const float*)d_in[13], (const float*)d_in[17]};
  const float* b2[3]    = {(const float*)d_in[10], (const float*)d_in[14], (const float*)d_in[18]};
  const float* fc1_w  = (const float*)d_in[19];
  const float* fc1_b  = (const float*)d_in[20];
  const float* fc2_w  = (const float*)d_in[21];
  const float* fc2_b  = (const float*)d_in[22];
  const float* conv_w = (const float*)d_in[23];
  const float* conv_b = (const float*)d_in[24];

  // workspace layout (floats): ~23.6 MB total
  float* ws     = (float*)d_ws;
  float* ahat   = ws;                                 // 3*NP*NP
  float* deg    = ahat + (size_t)3 * NP * NP;         // 3*NP
  float* dinv   = deg  + 3 * NP;                      // 3*NP
  float* pool   = dinv + 3 * NP;                      // 8
  float* a_att  = pool + 8;                           // 8
  float* H      = a_att + 8;                          // 6*NP*FDIM
  float* xw     = H + (size_t)6 * NP * FDIM;          // NP*FDIM
  float* x_pad  = xw + (size_t)NP * FDIM;             // NP*FDIM

  init_ws<<<1024, 256, 0, stream>>>(ahat, deg, pool);

  pad_dm1<<<(NP * FDIM + 255) / 256, 256, 0, stream>>>(dm1, x_pad);

  deg_accum<<<(3 * NEDGE + 255) / 256, 256, 0, stream>>>(
      edges[0], ew[0], edges[1], ew[1], edges[2], ew[2], deg);

  compute_dinv<<<(3 * NP + 255) / 256, 256, 0, stream>>>(deg, dinv);

  build_ahat<<<(3 * (NEDGE + N_DRUGS) + 255) / 256, 256, 0, stream>>>(
      edges[0], ew[0], edges[1], ew[1], edges[2], ew[2], dinv, ahat);

  const int m_tiles  = NP / 16;        // 37 (16-row strips)
  const int n_wtiles = FDIM / 64;      // 16 (64-col strips per wave)
  const int waves    = m_tiles * n_wtiles;            // 592 waves
  const int blocks   = (waves * 32 + 255) / 256;      // 74 blocks of 8 waves

  for (int t = 0; t < 3; ++t) {
    float* Ah = ahat + (size_t)t * NP * NP;
    float* h1 = H + (size_t)(2 * t)     * NP * FDIM;
    float* h2 = H + (size_t)(2 * t + 1) * NP * FDIM;

    // xw = x_pad @ W1               [592x1024 @ 1024x1024] (pad row -> 0)
    gemm_wmma_f32<<<blocks, 256, 0, stream>>>(x_pad, FDIM, W1[t], FDIM,
                                              xw, FDIM, FDIM, n_wtiles, m_tiles,
                                              nullptr, 0);
    // h1 = relu(Ahat @ xw + b1)     [592x592 @ 592x1024]
    gemm_wmma_f32<<<blocks, 256, 0, stream>>>(Ah, NP, xw, FDIM,
                                              h1, FDIM, NP, n_wtiles, m_tiles,
                                              b1[t], 1);
    // xw = h1 @ W2
    gemm_wmma_f32<<<blocks, 256, 0, stream>>>(h1, FDIM, W2[t], FDIM,
                                              xw, FDIM, FDIM, n_wtiles, m_tiles,
                                              nullptr, 0);
    // h2 = relu(Ahat @ xw + b2)
    gemm_wmma_f32<<<blocks, 256, 0, stream>>>(Ah, NP, xw, FDIM,
                                              h2, FDIM, NP, n_wtiles, m_tiles,
                                              b2[t], 1);
  }

  dim3 pgrid(64, 6);
  pool_reduce<<<pgrid, 256, 0, stream>>>(H, pool);
  attn_kernel<<<1, 32, 0, stream>>>(pool, fc1_w, fc1_b, fc2_w, fc2_b, a_att);
  final_kernel<<<(N_DRUGS * FDIM + 255) / 256, 256, 0, stream>>>(
      H, a_att, conv_w, conv_b, (float*)d_out);
}